// AdditiveAttention_53523882443415
// MI455X (gfx1250) — compile-verified
//
#include <hip/hip_runtime.h>
#include <hip/hip_bf16.h>

typedef __attribute__((ext_vector_type(2))) float v2f;
typedef __attribute__((ext_vector_type(8))) float v8f;

// gfx1250 hardware transcendental tanh (TRANS32 op). Inline asm guarantees we
// get the single-cycle-class v_tanh_f32 rather than the ocml software tanh.
static __device__ __forceinline__ float htanh(float x) {
    float r;
    asm("v_tanh_f32 %0, %1" : "=v"(r) : "v"(x));
    return r;
}
#define TANHF(x) htanh(x)

// ---------------------------------------------------------------------------
// Kernel 1/2: C[M,256] = X[M,256] @ W[256,256]^T + bias   (fp32 WMMA 16x16x4)
// One wave per 16x16 output tile; K-loop of 64 WMMAs.
// A frag: lane<16 -> X[m0+lane][k+0,k+1]; lane>=16 -> X[m0+lane-16][k+2,k+3]
// B frag: B[k][n] = W[n][k]: lane<16 -> W[n0+lane][k+0,k+1]; lane>=16 -> k+2,k+3
// ---------------------------------------------------------------------------
__global__ __launch_bounds__(128)
void proj_gemm_kernel(const float* __restrict__ X, const float* __restrict__ W,
                      const float* __restrict__ bias, float* __restrict__ C,
                      int M) {
    const int K = 256, N = 256;
    int wave = threadIdx.x >> 5;
    int lane = threadIdx.x & 31;
    int tile = blockIdx.x * 4 + wave;
    int tilesN = N >> 4;                 // 16
    int m0 = (tile / tilesN) << 4;
    int n0 = (tile % tilesN) << 4;
    if (m0 >= M) return;

    int ln = lane & 15;
    int kh = (lane >> 4) << 1;           // 0 or 2

    v8f c = {0.f,0.f,0.f,0.f,0.f,0.f,0.f,0.f};
    const float* xrow = X + (size_t)(m0 + ln) * K + kh;
    const float* wrow = W + (size_t)(n0 + ln) * K + kh;
    for (int k = 0; k < K; k += 4) {
        v2f a = *(const v2f*)(xrow + k);
        v2f b = *(const v2f*)(wrow + k);
        c = __builtin_amdgcn_wmma_f32_16x16x4_f32(false, a, false, b,
                                                  (short)0, c, false, false);
    }
    float bv = bias[n0 + ln];
    int mbase = m0 + ((lane >> 4) << 3); // +0 (lanes 0-15) or +8 (lanes 16-31)
    float* crow = C + (size_t)mbase * N + (n0 + ln);
#pragma unroll
    for (int i = 0; i < 8; ++i) crow[(size_t)i * N] = c[i] + bv;
}

// ---------------------------------------------------------------------------
// Kernel 3: scores + softmax -> normalized weights [B,T,S]
// Workgroup = (b, tile of 4 t-rows), 256 threads.
// ---------------------------------------------------------------------------
#define TT 4
#define ST 64
#define KPAD 260   // 256 + 4 dwords padding: kills 1KB-stride bank conflicts

__global__ __launch_bounds__(256)
void scores_softmax_kernel(const float* __restrict__ qp, const float* __restrict__ kp,
                           const float* __restrict__ va, const float* __restrict__ ba,
                           float* __restrict__ wts) {
    __shared__ float q_s[TT][KPAD];
    __shared__ float k_s[ST][KPAD];
    __shared__ float v_s[256];
    __shared__ float sc_s[TT][512];

    const int tid = threadIdx.x;
    const int b  = blockIdx.x / (128 / TT);
    const int t0 = (blockIdx.x % (128 / TT)) * TT;
    const float ba0 = ba[0];

    // stage q tile (4x256) as float4: 256 transfers == blockDim
    {
        int t   = tid >> 6;
        int col = (tid & 63) << 2;
        *(float4*)&q_s[t][col] =
            *(const float4*)&qp[((size_t)(b * 128 + t0 + t)) * 256 + col];
    }
    v_s[tid] = va[tid];

    const int t  = tid >> 6;     // 0..3
    const int sl = tid & 63;     // 0..63

    for (int st = 0; st < 512; st += ST) {
        __syncthreads();
        // stage k tile (64x256) as float4: 4096 transfers, 16 per thread
        for (int i = tid; i < ST * 64; i += 256) {
            int row = i >> 6;
            int col = (i & 63) << 2;
            *(float4*)&k_s[row][col] =
                *(const float4*)&kp[((size_t)(b * 512 + st + row)) * 256 + col];
        }
        __syncthreads();

        float acc = 0.f;
#pragma unroll 4
        for (int a = 0; a < 256; a += 4) {
            float4 qv = *(const float4*)&q_s[t][a];
            float4 kv = *(const float4*)&k_s[sl][a];
            float4 vv = *(const float4*)&v_s[a];
            acc += vv.x * TANHF(qv.x + kv.x);
            acc += vv.y * TANHF(qv.y + kv.y);
            acc += vv.z * TANHF(qv.z + kv.z);
            acc += vv.w * TANHF(qv.w + kv.w);
        }
        sc_s[t][st + sl] = acc + ba0;
    }
    __syncthreads();

    // softmax: wave w (w < 4) handles row w (512 values, 16 per lane)
    int wv   = tid >> 5;
    int lane = tid & 31;
    if (wv < TT) {
        float vals[16];
        float m = -1e30f;
#pragma unroll
        for (int i = 0; i < 16; ++i) {
            vals[i] = sc_s[wv][lane + (i << 5)];
            m = fmaxf(m, vals[i]);
        }
        for (int off = 16; off > 0; off >>= 1)
            m = fmaxf(m, __shfl_xor(m, off, 32));
        float sum = 0.f;
#pragma unroll
        for (int i = 0; i < 16; ++i) { vals[i] = __expf(vals[i] - m); sum += vals[i]; }
        for (int off = 16; off > 0; off >>= 1)
            sum += __shfl_xor(sum, off, 32);
        float inv = 1.f / sum;
        float* wrow = wts + ((size_t)(b * 128 + t0 + wv)) * 512;
#pragma unroll
        for (int i = 0; i < 16; ++i) wrow[lane + (i << 5)] = vals[i] * inv;
    }
}

// ---------------------------------------------------------------------------
// Kernel 4: out[b] = weights[b] (128x512) @ values[b] (512x256)  (fp32 WMMA)
// One wave per 16x16 tile; 128 tiles per batch.
// ---------------------------------------------------------------------------
__global__ __launch_bounds__(128)
void out_gemm_kernel(const float* __restrict__ wts, const float* __restrict__ values,
                     float* __restrict__ out) {
    int wave = threadIdx.x >> 5;
    int lane = threadIdx.x & 31;
    int gtile = blockIdx.x * 4 + wave;
    int b    = gtile >> 7;       // 128 tiles per batch
    int tile = gtile & 127;
    int m0 = (tile >> 4) << 4;   // 8 M-tiles
    int n0 = (tile & 15) << 4;   // 16 N-tiles

    int ln = lane & 15;
    int kh = (lane >> 4) << 1;

    v8f c = {0.f,0.f,0.f,0.f,0.f,0.f,0.f,0.f};
    const float* arow = wts + ((size_t)(b * 128 + m0 + ln)) * 512 + kh;
    const float* bcol = values + ((size_t)(b * 512 + kh)) * 256 + (n0 + ln);
    for (int k = 0; k < 512; k += 4) {
        v2f a = *(const v2f*)(arow + k);
        v2f bb;
        bb.x = bcol[(size_t)k * 256];
        bb.y = bcol[(size_t)k * 256 + 256];
        c = __builtin_amdgcn_wmma_f32_16x16x4_f32(false, a, false, bb,
                                                  (short)0, c, false, false);
    }
    int mbase = m0 + ((lane >> 4) << 3);
    float* crow = out + ((size_t)(b * 128 + mbase)) * 256 + (n0 + ln);
#pragma unroll
    for (int i = 0; i < 8; ++i) crow[(size_t)i * 256] = c[i];
}

// ---------------------------------------------------------------------------
extern "C" void kernel_launch(void* const* d_in, const int* in_sizes, int n_in,
                              void* d_out, int out_size, void* d_ws, size_t ws_size,
                              hipStream_t stream) {
    const float* queries = (const float*)d_in[0]; // [8,128,256]
    const float* keys    = (const float*)d_in[1]; // [8,512,256]
    const float* values  = (const float*)d_in[2]; // [8,512,256]
    const float* W_q     = (const float*)d_in[3]; // [256,256]
    const float* b_q     = (const float*)d_in[4]; // [256]
    const float* W_k     = (const float*)d_in[5]; // [256,256]
    const float* b_k     = (const float*)d_in[6]; // [256]
    const float* v_a     = (const float*)d_in[7]; // [1,256]
    const float* b_a     = (const float*)d_in[8]; // [1]

    float* ws  = (float*)d_ws;
    float* qp  = ws;                          // 1024*256   = 262144 floats
    float* kp  = ws + 262144;                 // 4096*256   = 1048576 floats
    float* wts = ws + 262144 + 1048576;       // 1024*512   = 524288 floats

    // q projection: M=1024 -> 1024 tiles -> 256 blocks of 4 waves
    proj_gemm_kernel<<<256, 128, 0, stream>>>(queries, W_q, b_q, qp, 1024);
    // k projection: M=4096 -> 4096 tiles -> 1024 blocks
    proj_gemm_kernel<<<1024, 128, 0, stream>>>(keys, W_k, b_k, kp, 4096);
    // scores + softmax: 8 batches * 32 t-tiles = 256 workgroups
    scores_softmax_kernel<<<256, 256, 0, stream>>>(qp, kp, v_a, b_a, wts);
    // output GEMM: 8 * 128 tiles / 4 waves = 256 blocks
    out_gemm_kernel<<<256, 128, 0, stream>>>(wts, values, (float*)d_out);
}